// TreeLSTMClassifier_1391569404595
// MI455X (gfx1250) — compile-verified
//
#include <hip/hip_runtime.h>
#include <hip/hip_bf16.h>

typedef __attribute__((ext_vector_type(2))) float v2f;
typedef __attribute__((ext_vector_type(8))) float v8f;

#define VOCAB 100000
#define E 300
#define H 300
#define HC 600       // 2*H
#define NP 1500      // 5*H
#define Bsz 128
#define Tn 128
#define Mrows (Bsz*Tn)   // 16384
#define NSTEPS (2*Tn-1)  // 255
#define PAD 1

#define K1_NT 19     // n-tiles for H=300
#define K1_KS 75     // k-steps for K=300
#define K2_NT 94     // n-tiles for NP=1500
#define K2_KS 150    // k-steps for K=600

__device__ __forceinline__ float sigmoidf_(float x) {
    return 1.0f / (1.0f + expf(-x));
}

__device__ __forceinline__ v8f wmma_f32(v2f a, v2f b, v8f c) {
    return __builtin_amdgcn_wmma_f32_16x16x4_f32(
        false, a, false, b, (short)0, c, false, false);
}

// ---------------------------------------------------------------------------
// Pack a row-major (K x N) weight matrix into WMMA B-fragment order:
//   packed[(nt*KS + ks)*32 + lane] = { W[ka][n], W[ka+1][n] }  (v2f)
// with n = nt*16 + (lane&15) clamped to N-1 (edge tiles masked at store time
// in the GEMM), ka = ks*4 + 2*(lane>>4).
// ---------------------------------------------------------------------------
__global__ void __launch_bounds__(256)
pack_b_kernel(const float* __restrict__ W, float* __restrict__ packed,
              int K, int N, int NT)
{
    const int KS = K >> 2;
    const int total = NT * KS * 32;
    const int idx = blockIdx.x * blockDim.x + threadIdx.x;
    if (idx >= total) return;
    const int lane = idx & 31;
    const int ks   = (idx >> 5) % KS;
    const int nt   = (idx >> 5) / KS;
    int n = nt * 16 + (lane & 15);
    if (n >= N) n = N - 1;                       // clamp; GEMM predicates stores
    const int ka = ks * 4 + ((lane >> 4) << 1);
    packed[(size_t)idx * 2 + 0] = W[(size_t)ka * N + n];
    packed[(size_t)idx * 2 + 1] = W[(size_t)(ka + 1) * N + n];
}

// ---------------------------------------------------------------------------
// Kernel 1: leaf transform.  buffers[b*T+t] = [h(300), c(300)]
//   c = emb @ W_proj + b_proj ; h = tanh(c) * sigmoid(emb @ W_gate + b_gate)
// GEMM M=16384, N=300 (19 tiles, edge-masked at store), K=300 (75 k-steps).
// One wave per (mtile, ntile); both proj & gate accumulated from one A frag.
// Note: embed_table[PAD] is all-zero by construction, so no mask needed.
// ---------------------------------------------------------------------------
__global__ void __launch_bounds__(128)
leaf_kernel(const int* __restrict__ tokens, const float* __restrict__ table,
            const v2f* __restrict__ packP, const float* __restrict__ bp,
            const v2f* __restrict__ packG, const float* __restrict__ bg,
            float* __restrict__ buffers)
{
    const int wave = (blockIdx.x * blockDim.x + threadIdx.x) >> 5;
    const int lane = threadIdx.x & 31;
    const int nt = wave % K1_NT;
    const int mt = wave / K1_NT;         // 0..1023 exactly (grid sized to fit)

    const int m0 = mt * 16;
    const int kofs = (lane >> 4) * 2;    // 0 or 2

    const int m   = m0 + (lane & 15);
    const int tok = tokens[m];
    const float* __restrict__ arow = table + (long)tok * E;

    const v2f* __restrict__ bP = packP + (size_t)nt * K1_KS * 32 + lane;
    const v2f* __restrict__ bG = packG + (size_t)nt * K1_KS * 32 + lane;

    v8f accP = {};
    v8f accG = {};
    for (int ks = 0; ks < K1_KS; ++ks) {
        const v2f a = *(const v2f*)(arow + ks * 4 + kofs);
        accP = wmma_f32(a, bP[ks * 32], accP);
        accG = wmma_f32(a, bG[ks * 32], accG);
    }

    const int n = nt * 16 + (lane & 15);
    if (n < H) {
        const float biasP = bp[n];
        const float biasG = bg[n];
        const int mbase = m0 + ((lane >> 4) ? 8 : 0);
        #pragma unroll
        for (int i = 0; i < 8; ++i) {
            const long mm = mbase + i;
            const float c = accP[i] + biasP;
            const float g = accG[i] + biasG;
            const float h = tanhf(c) * sigmoidf_(g);
            buffers[mm * HC + n]     = h;
            buffers[mm * HC + H + n] = c;
        }
    }
}

// ---------------------------------------------------------------------------
// Kernel 2 (per scan step): proj = [h_l | h_r] @ W_reduce + b_reduce
// A row b, col j:  j<300 -> stack[b][max(sp-2,0)][j]   (h of left)
//                  j>=300-> stack[b][max(sp-1,0)][j-300] (h of right)
// GEMM M=128 (8 tiles), N=1500 (94 tiles, edge-masked), K=600 (150 k-steps,
// split into two uniform halves so the hl/hr select is hoisted out).
// ---------------------------------------------------------------------------
__global__ void __launch_bounds__(128)
reduce_gemm_kernel(const float* __restrict__ stack, const int* __restrict__ sp,
                   const v2f* __restrict__ packR, const float* __restrict__ br,
                   float* __restrict__ P)
{
    const int wave = (blockIdx.x * blockDim.x + threadIdx.x) >> 5;
    const int lane = threadIdx.x & 31;
    const int nt = wave % K2_NT;
    const int mt = wave / K2_NT;         // 0..7 exactly

    const int m0 = mt * 16;
    const int kofs = (lane >> 4) * 2;

    const int b = m0 + (lane & 15);
    const int s = sp[b];
    const float* __restrict__ base = stack + (long)b * (Tn * HC);
    const float* __restrict__ hl = base + (long)max(s - 2, 0) * HC;
    const float* __restrict__ hr = base + (long)max(s - 1, 0) * HC;

    const v2f* __restrict__ bR = packR + (size_t)nt * K2_KS * 32 + lane;

    v8f acc = {};
    for (int ks = 0; ks < 75; ++ks) {                    // K = 0..299 -> h_l
        const v2f a = *(const v2f*)(hl + ks * 4 + kofs);
        acc = wmma_f32(a, bR[ks * 32], acc);
    }
    for (int ks = 75; ks < K2_KS; ++ks) {                // K = 300..599 -> h_r
        const v2f a = *(const v2f*)(hr + (ks - 75) * 4 + kofs);
        acc = wmma_f32(a, bR[ks * 32], acc);
    }

    const int n = nt * 16 + (lane & 15);
    if (n < NP) {
        const float bias = br[n];
        const int mbase = m0 + ((lane >> 4) ? 8 : 0);
        #pragma unroll
        for (int i = 0; i < 8; ++i) {
            const long bb = mbase + i;
            P[bb * NP + n] = acc[i] + bias;
        }
    }
}

// ---------------------------------------------------------------------------
// Kernel 3 (per scan step): LSTM combine + shift/reduce stack update.
// One thread per (b, n), n in [0,300). Double-buffered sp/bp (no races:
// each stack address touched by exactly one thread, read-before-write).
// ---------------------------------------------------------------------------
__global__ void __launch_bounds__(256)
combine_kernel(const float* __restrict__ P, float* __restrict__ stack,
               const float* __restrict__ buffers, const int* __restrict__ tr_row,
               const int* __restrict__ sp_in, const int* __restrict__ bp_in,
               int* __restrict__ sp_out, int* __restrict__ bp_out)
{
    const int idx = blockIdx.x * blockDim.x + threadIdx.x;
    if (idx >= Bsz * H) return;
    const int b = idx / H;
    const int n = idx % H;

    const int s  = sp_in[b];
    const int bp = bp_in[b];
    const bool is_shift = (tr_row[b] == 0);

    float* __restrict__ base = stack + (long)b * (Tn * HC);
    const float c_l = base[(long)max(s - 2, 0) * HC + H + n];
    const float c_r = base[(long)max(s - 1, 0) * HC + H + n];

    const float* __restrict__ prow = P + (long)b * NP;
    const float ii = sigmoidf_(prow[n]);
    const float fl = sigmoidf_(prow[H + n]);
    const float fr = sigmoidf_(prow[2 * H + n]);
    const float gg = tanhf(prow[3 * H + n]);
    const float oo = sigmoidf_(prow[4 * H + n]);
    const float c = ii * gg + fl * c_l + fr * c_r;
    const float h = oo * tanhf(c);

    const int bpc = max(bp, 0);
    const float* __restrict__ shrow = buffers + ((long)b * Tn + bpc) * HC;
    const float nh = is_shift ? shrow[n]     : h;
    const float nc = is_shift ? shrow[H + n] : c;

    const int pos = is_shift ? s : max(s - 2, 0);
    float* __restrict__ dst = base + (long)pos * HC;
    dst[n]     = nh;
    dst[H + n] = nc;

    if (n == 0) {
        sp_out[b] = is_shift ? (s + 1) : (s - 1);
        bp_out[b] = is_shift ? (bp - 1) : bp;
    }
}

// ---------------------------------------------------------------------------
// Kernel 4: output head. out[b][o] = root_h(b) . W_out[:,o] + b_out[o]
// ---------------------------------------------------------------------------
__global__ void __launch_bounds__(128)
head_kernel(const float* __restrict__ stack, const int* __restrict__ sp,
            const float* __restrict__ Wout, const float* __restrict__ bout,
            float* __restrict__ out)
{
    const int idx = blockIdx.x * blockDim.x + threadIdx.x;
    if (idx >= Bsz * 5) return;
    const int b = idx / 5;
    const int o = idx % 5;
    const int s = sp[b];
    const float* __restrict__ root = stack + ((long)b * Tn + max(s - 1, 0)) * HC;
    float acc = bout[o];
    for (int k = 0; k < H; ++k) acc += root[k] * Wout[k * 5 + o];
    out[idx] = acc;
}

__global__ void init_kernel(int* __restrict__ sp0, int* __restrict__ bp0)
{
    const int b = threadIdx.x;
    if (b < Bsz) { sp0[b] = 0; bp0[b] = Tn - 1; }
}

// ---------------------------------------------------------------------------
extern "C" void kernel_launch(void* const* d_in, const int* in_sizes, int n_in,
                              void* d_out, int out_size, void* d_ws, size_t ws_size,
                              hipStream_t stream) {
    const int*   tokens      = (const int*)  d_in[0];
    const int*   transitions = (const int*)  d_in[1];   // (255, 128)
    const float* embed_table = (const float*)d_in[2];
    const float* W_proj      = (const float*)d_in[3];
    const float* b_proj      = (const float*)d_in[4];
    const float* W_gate      = (const float*)d_in[5];
    const float* b_gate      = (const float*)d_in[6];
    const float* W_reduce    = (const float*)d_in[7];
    const float* b_reduce    = (const float*)d_in[8];
    const float* W_out       = (const float*)d_in[9];
    const float* b_out       = (const float*)d_in[10];
    float* out = (float*)d_out;

    char* ws = (char*)d_ws;
    const size_t SZ_BUF   = (size_t)Mrows * HC * sizeof(float);        // 39.3 MB
    const size_t SZ_STACK = (size_t)Bsz * Tn * HC * sizeof(float);     // 39.3 MB
    const size_t SZ_P     = (size_t)Bsz * NP * sizeof(float);          // 0.77 MB
    const size_t SZ_PKR   = (size_t)K2_NT * K2_KS * 32 * 2 * sizeof(float); // 3.6 MB
    const size_t SZ_PK1   = (size_t)K1_NT * K1_KS * 32 * 2 * sizeof(float); // 365 KB

    float* buffers = (float*)(ws);
    float* stack   = (float*)(ws + SZ_BUF);
    float* P       = (float*)(ws + SZ_BUF + SZ_STACK);
    float* packRf  = (float*)(ws + SZ_BUF + SZ_STACK + SZ_P);
    float* packPf  = (float*)(ws + SZ_BUF + SZ_STACK + SZ_P + SZ_PKR);
    float* packGf  = (float*)(ws + SZ_BUF + SZ_STACK + SZ_P + SZ_PKR + SZ_PK1);
    int*   spbuf   = (int*)  (ws + SZ_BUF + SZ_STACK + SZ_P + SZ_PKR + 2 * SZ_PK1);
    int*   bpbuf   = spbuf + 2 * Bsz;

    init_kernel<<<1, 128, 0, stream>>>(spbuf, bpbuf);

    // Pack weights into WMMA B-fragment order (once per launch).
    {
        const int totR = K2_NT * K2_KS * 32;
        pack_b_kernel<<<(totR + 255) / 256, 256, 0, stream>>>(W_reduce, packRf, HC, NP, K2_NT);
        const int tot1 = K1_NT * K1_KS * 32;
        pack_b_kernel<<<(tot1 + 255) / 256, 256, 0, stream>>>(W_proj, packPf, E, H, K1_NT);
        pack_b_kernel<<<(tot1 + 255) / 256, 256, 0, stream>>>(W_gate, packGf, E, H, K1_NT);
    }

    // Leaf transform: 1024 mtiles * 19 ntiles waves, 4 waves/block
    {
        const int waves = 1024 * K1_NT;
        leaf_kernel<<<waves / 4, 128, 0, stream>>>(
            tokens, embed_table, (const v2f*)packPf, b_proj,
            (const v2f*)packGf, b_gate, buffers);
    }

    // Sequential shift/reduce scan: 255 steps, GEMM + combine per step
    for (int s = 0; s < NSTEPS; ++s) {
        const int* sp_in  = spbuf + (s & 1) * Bsz;
        const int* bp_in  = bpbuf + (s & 1) * Bsz;
        int* sp_out = spbuf + ((s + 1) & 1) * Bsz;
        int* bp_out = bpbuf + ((s + 1) & 1) * Bsz;

        const int waves = 8 * K2_NT;   // 752 waves, 188 blocks of 4 waves
        reduce_gemm_kernel<<<waves / 4, 128, 0, stream>>>(
            stack, sp_in, (const v2f*)packRf, b_reduce, P);

        combine_kernel<<<(Bsz * H + 255) / 256, 256, 0, stream>>>(
            P, stack, buffers, transitions + (long)s * Bsz,
            sp_in, bp_in, sp_out, bp_out);
    }

    head_kernel<<<(Bsz * 5 + 127) / 128, 128, 0, stream>>>(
        stack, spbuf + (NSTEPS & 1) * Bsz, W_out, b_out, out);
}